// IDWT1D_40303973105781
// MI455X (gfx1250) — compile-verified
//
#include <hip/hip_runtime.h>
#include <stdint.h>

// IDWT-1D along axis 2 of x:(16,256,64,256) f32.
// out[..,2m,  c] = lp1*a[m] + lp3*a[(m+1)%32] + hp0*d[(m-1)%32] + hp2*d[m]
// out[..,2m+1,c] = lp0*a[m] + lp2*a[(m+1)%32] + hp1*d[m]        + hp3*d[(m+1)%32]
// a = x rows 0..31, d = x rows 32..63 of the N axis.
//
// Memory-bound (512 MiB traffic, ~23 us @ 23.3 TB/s). Strategy:
//  - one workgroup per (b,h) slice: slice input is a contiguous 64 KB block
//  - async-stage it to LDS with global_load_async_to_lds_b128 (ASYNCcnt)
//  - compute from LDS (wrap rows reused for free), NT b128 stores to global

typedef float v4f __attribute__((ext_vector_type(4)));

#define SLICE_C4   64                    // 256 floats per row = 64 float4
#define SLICE_N    64                    // rows per slice
#define SLICE_V4   (SLICE_N * SLICE_C4)  // 4096 float4 = 64 KB

__global__ __launch_bounds__(256)
void idwt1d_kernel(const float* __restrict__ x,
                   const float* __restrict__ lp,
                   const float* __restrict__ hp,
                   float* __restrict__ out)
{
    __shared__ v4f tile[SLICE_V4];       // 64 KB: rows 0..31 approx, 32..63 detail

    const uint32_t tid   = threadIdx.x;
    const uint64_t slice = blockIdx.x;
    const uint64_t base  = slice * (uint64_t)(SLICE_N * 256);   // float index

    // ---- async stage: contiguous 64 KB (slice) -> LDS -----------------
    {
        const uint64_t saddr = (uint64_t)(uintptr_t)(x + base);
        const uint32_t lds0  = (uint32_t)(uintptr_t)(&tile[0]); // low 32b = LDS byte offset
#pragma unroll
        for (int i = 0; i < 16; ++i) {
            uint32_t idx    = tid + i * 256;     // float4 index within slice
            uint32_t voff   = idx * 16;          // byte offset (global & LDS identical)
            uint32_t ldsoff = lds0 + voff;
            asm volatile("global_load_async_to_lds_b128 %0, %1, %2"
                         :: "v"(ldsoff), "v"(voff), "s"(saddr)
                         : "memory");
        }
        asm volatile("s_wait_asynccnt 0x0" ::: "memory");
    }
    __syncthreads();

    const float l0 = lp[0], l1 = lp[1], l2 = lp[2], l3 = lp[3];
    const float h0 = hp[0], h1 = hp[1], h2 = hp[2], h3 = hp[3];

    const uint32_t cq = tid & 63;        // float4 column
    const uint32_t mg = tid >> 6;        // 0..3
    const uint32_t m0 = mg * 8;          // this thread: m in [m0, m0+8)

    const v4f* ap = &tile[0];                 // approx rows [0,32)
    const v4f* dp = &tile[32 * SLICE_C4];     // detail rows [0,32)

    v4f a0  = ap[ m0            * 64 + cq];
    v4f dm1 = dp[((m0 + 31) & 31) * 64 + cq];
    v4f d0  = dp[ m0            * 64 + cq];

    float* const outp = out + base;
#pragma unroll
    for (uint32_t j = 0; j < 8; ++j) {
        const uint32_t m   = m0 + j;
        const uint32_t mp1 = (m + 1) & 31;
        v4f a1 = ap[mp1 * 64 + cq];
        v4f d1 = dp[mp1 * 64 + cq];

        v4f oe, oo;
#pragma unroll
        for (int c = 0; c < 4; ++c) {
            float e = l1 * a0[c];
            e = fmaf(l3, a1[c],  e);
            e = fmaf(h0, dm1[c], e);
            e = fmaf(h2, d0[c],  e);
            float o = l0 * a0[c];
            o = fmaf(l2, a1[c], o);
            o = fmaf(h1, d0[c], o);
            o = fmaf(h3, d1[c], o);
            oe[c] = e;
            oo[c] = o;
        }
        // streaming output: non-temporal b128 stores (keep L2 for the input stream)
        __builtin_nontemporal_store(oe, (v4f*)(outp + (uint64_t)(2 * m)     * 256 + cq * 4));
        __builtin_nontemporal_store(oo, (v4f*)(outp + (uint64_t)(2 * m + 1) * 256 + cq * 4));

        a0  = a1;   // rolling reuse: 2 LDS loads per 2 output rows
        dm1 = d0;
        d0  = d1;
    }
}

extern "C" void kernel_launch(void* const* d_in, const int* in_sizes, int n_in,
                              void* d_out, int out_size, void* d_ws, size_t ws_size,
                              hipStream_t stream) {
    const float* x  = (const float*)d_in[0];
    const float* lp = (const float*)d_in[1];
    const float* hp = (const float*)d_in[2];
    float* out      = (float*)d_out;

    const int nslices = out_size / (SLICE_N * 256);   // 16*256 = 4096
    if (nslices <= 0) return;
    idwt1d_kernel<<<dim3((uint32_t)nslices), dim3(256), 0, stream>>>(x, lp, hp, out);
}